// TensorConvLayerInput_91259465105624
// MI455X (gfx1250) — compile-verified
//
#include <hip/hip_runtime.h>
#include <cstdint>

typedef float v2f __attribute__((ext_vector_type(2)));
typedef float v8f __attribute__((ext_vector_type(8)));
typedef __attribute__((address_space(3))) float       ldsf_t;
typedef __attribute__((address_space(3))) const v2f   lds_cv2f;

// ---- problem constants (from reference shapes) ----
#define MTOT   147852      // 4*111*111*3   (b,i,j,e)
#define NTOT   500         // 4*125         (g,p,q,r)
#define NKB    18          // K=288 in blocks of 16
// x strides (elements): [b,c,h,w,e,f] = (4,8,224,224,3,4)
#define XB 4816896
#define XC 602112
#define XH 2688
#define XW 12
// weight strides (elements): [g,c,kh,kw,f,p,q,r] = (4,8,3,3,4,5,5,5)
#define WGs 36000
#define WCs 4500
#define WKs 1500
#define WLs 500
#define WFs 125
// out strides (elements): [b,g,i,j,e,p,q,r]
#define OBs 18481500u
#define OGs 4620375u
#define MREM 36963         // 111*111*3 (rows per batch)

#define M_BLK 256
#define N_BLK 64
#define LDS_STRIDE 20                       // pad: 16B-aligned rows, conflict-free frag reads
#define A_FLOATS (M_BLK * LDS_STRIDE)       // 5120
#define B_FLOATS (N_BLK * LDS_STRIDE)       // 1280
#define BUF_FLOATS (A_FLOATS + B_FLOATS)    // 6400

// ---- compile-time (c,k,l) group-offset tables in BYTES: G = c*9+k*3+l ----
struct GroupTab { int x[72]; int w[72]; };
static constexpr GroupTab makeTab() {
    GroupTab t{};
    for (int G = 0; G < 72; ++G) {
        const int c = G / 9, rm = G % 9, k = rm / 3, l = rm % 3;
        t.x[G] = 4 * (c * XC + k * XH + l * XW);      // byte offset into x
        t.w[G] = 4 * (c * WCs + k * WKs + l * WLs);   // byte offset into weights
    }
    return t;
}
__device__ __constant__ GroupTab gTab = makeTab();

__global__ __launch_bounds__(256)
void tconv_wmma_f32(const float* __restrict__ x,
                    const float* __restrict__ w,
                    float* __restrict__ out)
{
    __shared__ float smem[2 * BUF_FLOATS];   // 51,200 B, double buffered
    ldsf_t* sm3 = (ldsf_t*)smem;             // AS(3) view: DS addressing + address escape

    const int tid = threadIdx.x;
    const int m0  = blockIdx.x * M_BLK;
    const int n0  = blockIdx.y * N_BLK;

    // ---- A duty: one im2col row per thread (clamped if out of range) ----
    const int m = m0 + tid;
    uint32_t aByte = 0;                      // byte offset of row base in x
    if (m < MTOT) {
        const int e  = m % 3;    const int t1 = m / 3;
        const int jj = t1 % 111; const int t2 = t1 / 111;
        const int ii = t2 % 111; const int bb = t2 / 111;
        aByte = (uint32_t)(4 * (bb * XB + (2 * ii) * XH + (2 * jj) * XW + e * 4));
    }

    // ---- B duty: one N column, 4 K rows; f == bRow ----
    const int bCol = tid & 63;
    const int bRow = tid >> 6;               // 0..3 == f
    const int nCol = n0 + bCol;
    uint32_t bByte = (uint32_t)(4 * (bRow * WFs));  // clamped column base if OOB
    if (nCol < NTOT) {
        const int g = nCol / 125;
        bByte += (uint32_t)(4 * (g * WGs + (nCol - g * 125)));
    }

    // ---- loop-invariant LDS destination addresses (true LDS byte addresses) ----
    uint32_t aLdsAddr[2][4], bLdsAddr[2][4];
#pragma unroll
    for (int bf = 0; bf < 2; ++bf) {
#pragma unroll
        for (int i = 0; i < 4; ++i) {
            aLdsAddr[bf][i] = (uint32_t)(uintptr_t)
                (sm3 + bf * BUF_FLOATS + tid * LDS_STRIDE + i * 4);
            bLdsAddr[bf][i] = (uint32_t)(uintptr_t)
                (sm3 + bf * BUF_FLOATS + A_FLOATS + bCol * LDS_STRIDE + bRow + i * 4);
        }
    }

    // ---- async stage: global -> LDS via the CDNA5 async engine (ASYNCcnt) ----
    auto issueAsync = [&](int kb, int buf) {
        const int gb = kb * 4;               // uniform -> scalar table loads
#pragma unroll
        for (int i = 0; i < 4; ++i) {
            const uint32_t voff = aByte + (uint32_t)gTab.x[gb + i];
            asm volatile("global_load_async_to_lds_b128 %0, %1, %2"
                         :: "v"(aLdsAddr[buf][i]), "v"(voff), "s"(x)
                         : "memory");
        }
#pragma unroll
        for (int j = 0; j < 4; ++j) {
            const uint32_t voff = bByte + (uint32_t)gTab.w[gb + j];
            asm volatile("global_load_async_to_lds_b32 %0, %1, %2"
                         :: "v"(bLdsAddr[buf][j]), "v"(voff), "s"(w)
                         : "memory");
        }
    };

    // ---- compute mapping ----
    const int lane = tid & 31;
    const int wave = tid >> 5;
    const int lid  = lane & 15;
    const int half = lane >> 4;

    const v8f vzero = {0.f, 0.f, 0.f, 0.f, 0.f, 0.f, 0.f, 0.f};
    v8f acc0[4], acc1[4];
#pragma unroll
    for (int nt = 0; nt < 4; ++nt) { acc0[nt] = vzero; acc1[nt] = vzero; }

    auto compute = [&](int buf) {
        ldsf_t* s   = sm3 + buf * BUF_FLOATS;
        ldsf_t* ar0 = s + (wave * 32 + lid) * LDS_STRIDE;   // wave's M rows 0..15
        ldsf_t* ar1 = ar0 + 16 * LDS_STRIDE;                // M rows 16..31
        ldsf_t* bs  = s + A_FLOATS;
#pragma unroll
        for (int ks = 0; ks < 4; ++ks) {
            const int kk = ks * 4 + half * 2;  // lanes 0-15: K{0,1}; lanes 16-31: K{2,3}
            const v2f a0 = *(lds_cv2f*)(ar0 + kk);
            const v2f a1 = *(lds_cv2f*)(ar1 + kk);
#pragma unroll
            for (int nt = 0; nt < 4; ++nt) {
                const v2f b = *(lds_cv2f*)(bs + (nt * 16 + lid) * LDS_STRIDE + kk);
                acc0[nt] = __builtin_amdgcn_wmma_f32_16x16x4_f32(
                    false, a0, false, b, (short)0, acc0[nt], false, false);
                acc1[nt] = __builtin_amdgcn_wmma_f32_16x16x4_f32(
                    false, a1, false, b, (short)0, acc1[nt], false, false);
            }
        }
    };

    // ---- double-buffered K loop (unrolled x2 so buffer index is compile-time) ----
    issueAsync(0, 0);
    asm volatile("s_wait_asynccnt 0x0" ::: "memory");
    __syncthreads();
#pragma unroll 1
    for (int kb = 0; kb < NKB; kb += 2) {
        if (kb + 1 < NKB) issueAsync(kb + 1, 1);
        compute(0);
        asm volatile("s_wait_asynccnt 0x0" ::: "memory");
        __syncthreads();
        if (kb + 2 < NKB) issueAsync(kb + 2, 0);
        compute(1);
        asm volatile("s_wait_asynccnt 0x0" ::: "memory");
        __syncthreads();
    }

    // ---- epilogue: incremental row decomposition, hoisted column bases ----
    uint32_t nb[4];
    bool     nok[4];
#pragma unroll
    for (int nt = 0; nt < 4; ++nt) {
        const int nn = n0 + nt * 16 + lid;
        nok[nt] = (nn < NTOT);
        const int g = nn / 125;
        nb[nt] = (uint32_t)g * OGs + (uint32_t)(nn - g * 125);
    }

    const int mBase = m0 + wave * 32 + half * 8;  // C/D: VGPR v -> M = v + 8*half
#pragma unroll
    for (int s2 = 0; s2 < 2; ++s2) {              // s2=0 -> acc0 rows, s2=1 -> acc1 rows (+16)
        const int mb = mBase + s2 * 16;
        int bb  = mb / MREM;
        int rem = mb - bb * MREM;                 // (i*111+j)*3+e
        uint32_t bo = (uint32_t)bb * OBs;
#pragma unroll
        for (int v = 0; v < 8; ++v) {
            if (mb + v < MTOT) {
                const uint32_t ro = bo + (uint32_t)rem * 125u;
#pragma unroll
                for (int nt = 0; nt < 4; ++nt)
                    if (nok[nt]) out[ro + nb[nt]] = (s2 == 0 ? acc0[nt][v] : acc1[nt][v]);
            }
            if (++rem == MREM) { rem = 0; bo += OBs; }
        }
    }
}

extern "C" void kernel_launch(void* const* d_in, const int* in_sizes, int n_in,
                              void* d_out, int out_size, void* d_ws, size_t ws_size,
                              hipStream_t stream)
{
    (void)in_sizes; (void)n_in; (void)out_size; (void)d_ws; (void)ws_size;
    const float* x = (const float*)d_in[0];
    const float* w = (const float*)d_in[1];
    float* out = (float*)d_out;

    dim3 grid((MTOT + M_BLK - 1) / M_BLK,   // 578 M tiles
              (NTOT + N_BLK - 1) / N_BLK);  // 8 N tiles (covers 512 >= 500)
    dim3 block(256);
    hipLaunchKernelGGL(tconv_wmma_f32, grid, block, 0, stream, x, w, out);
}